// NodeDenoisingADMM_84018150244546
// MI455X (gfx1250) — compile-verified
//
#include <hip/hip_runtime.h>
#include <cstdint>
#include <cstddef>

// ---------------- problem constants (match reference) ----------------
#define GN     50000          // nodes
#define GF     64             // features
#define GNNZ   800000         // edges per operator
#define GL     4              // framelet operators
#define GITERS 5              // ADMM iterations
#define NF     ((size_t)GN * GF)        // 3,200,000 floats per [N,F] plane
#define LNF    ((size_t)GL * NF)        // 12,800,000 floats per [N,F,L] block
#define NF4    ((int)(NF / 4))          // float4 count of an [N,F] plane
#define LNF4   ((int)(LNF / 4))         // float4 count of an [N,F,L] block
#define NTILES (GNNZ / 32)              // 25,000 wave-tiles of 32 edges

// NOTE: T and Lam use interleaved layout [N, F, L] (L innermost):
//   element (n, f, l) lives at (n*GF + f)*GL + l
// so one float4 at index (n*GF + f) holds all 4 operators for that feature.

// ---------------- CDNA5 async-to-LDS staging ----------------
#if __has_builtin(__builtin_amdgcn_global_load_async_to_lds_b32)
#define USE_ASYNC_LDS 1
#else
#define USE_ASYNC_LDS 0
#endif

typedef __attribute__((address_space(1))) int glob_i32;
typedef __attribute__((address_space(3))) int lds_i32;

__device__ __forceinline__ void async_ld32(const void* g, void* l) {
#if USE_ASYNC_LDS
  __builtin_amdgcn_global_load_async_to_lds_b32(
      (glob_i32*)(void*)g, (lds_i32*)l, 0, 0);
#else
  (void)g; (void)l;
#endif
}

template <int C>
__device__ __forceinline__ void wait_asynccnt() {
#if __has_builtin(__builtin_amdgcn_s_wait_asynccnt)
  __builtin_amdgcn_s_wait_asynccnt(C);
#else
  asm volatile("s_wait_asynccnt %0" :: "i"(C) : "memory");
#endif
}

// relaxed agent-scope f32 atomic add; want: global_atomic_add_f32 (no return)
__device__ __forceinline__ void atomicAddF(float* p, float v) {
  __hip_atomic_fetch_add(p, v, __ATOMIC_RELAXED, __HIP_MEMORY_SCOPE_AGENT);
}

// ---- lowering probe: FIRST kernel in module so the disasm snippet shows
// exactly how atomicAddF lowers (native fadd vs CAS loop). Never launched.
__global__ void probe_atomic(float* p, float v) {
  atomicAddF(p, v);
}

// ---------------- per-edge bodies (32 lanes cooperate, 2 features/lane) ----
// pass 1: agg[row][f] += sum_l w_l * tmp[col][f][l]   (tmp interleaved)
__device__ __forceinline__ void edge_agg(int rr, int cc, float a0, float a1,
                                         float a2, float a3, int lane,
                                         const float4* __restrict__ tmp4,
                                         float* __restrict__ agg) {
  const size_t cbase = (size_t)cc * GF + 2 * lane;   // float4 index (n*GF+f)
  float4 A = tmp4[cbase];          // l0..l3 of feature f
  float4 B = tmp4[cbase + 1];      // l0..l3 of feature f+1
  float sx = a0 * A.x + a1 * A.y + a2 * A.z + a3 * A.w;
  float sy = a0 * B.x + a1 * B.y + a2 * B.z + a3 * B.w;
  float* dst = agg + (size_t)rr * GF + 2 * lane;
  atomicAddF(dst, sx);
  atomicAddF(dst + 1, sy);
}

// pass 2: wu[row][f][l] += w_l * uk[col][f]  -> 8 contiguous atomic adds
__device__ __forceinline__ void edge_wu(int rr, int cc, float a0, float a1,
                                        float a2, float a3, int lane,
                                        const float* __restrict__ uk,
                                        float* __restrict__ wu) {
  float2 u = *(const float2*)(uk + (size_t)cc * GF + 2 * lane);
  float* p = wu + ((size_t)rr * GF + 2 * lane) * GL;   // 32B contiguous run
  atomicAddF(p + 0, a0 * u.x);
  atomicAddF(p + 1, a1 * u.x);
  atomicAddF(p + 2, a2 * u.x);
  atomicAddF(p + 3, a3 * u.x);
  atomicAddF(p + 4, a0 * u.y);
  atomicAddF(p + 5, a1 * u.y);
  atomicAddF(p + 6, a2 * u.y);
  atomicAddF(p + 7, a3 * u.y);
}

// stage 32 edges (6 streams x 128B) of one tile into a wave-private LDS buffer
__device__ __forceinline__ void stage_tile(int tile, unsigned* buf, int lane,
                                           const int* __restrict__ rows,
                                           const int* __restrict__ cols,
                                           const float* __restrict__ wv) {
  const int e = tile * 32 + lane;
  async_ld32(rows + e,            buf + 0 * 32 + lane);
  async_ld32(cols + e,            buf + 1 * 32 + lane);
  async_ld32(wv + e,              buf + 2 * 32 + lane);
  async_ld32(wv + GNNZ + e,       buf + 3 * 32 + lane);
  async_ld32(wv + 2 * GNNZ + e,   buf + 4 * 32 + lane);
  async_ld32(wv + 3 * GNNZ + e,   buf + 5 * 32 + lane);
}

// ---------------- SpMM pass 2: WU_l = W_l @ Uk (interleaved dest) --------
__global__ __launch_bounds__(256) void scatter_wu(
    const float* __restrict__ wv, const int* __restrict__ rows,
    const int* __restrict__ cols, const float* __restrict__ uk,
    float* __restrict__ wu) {
  const int lane = threadIdx.x & 31;
  const int wib  = threadIdx.x >> 5;
  const int wave = blockIdx.x * (blockDim.x >> 5) + wib;
  const int nwav = gridDim.x * (blockDim.x >> 5);

#if USE_ASYNC_LDS
  __shared__ unsigned smem[8 * 2 * 6 * 32];
  unsigned* wbase = smem + wib * (2 * 6 * 32);
  int tile = wave;
  int parity = 0;
  if (tile < NTILES) stage_tile(tile, wbase, lane, rows, cols, wv);
  for (; tile < NTILES; tile += nwav) {
    const int nxt = tile + nwav;
    if (nxt < NTILES) {
      stage_tile(nxt, wbase + (parity ^ 1) * 6 * 32, lane, rows, cols, wv);
      wait_asynccnt<6>();
    } else {
      wait_asynccnt<0>();
    }
    asm volatile("" ::: "memory");
    const unsigned* buf = wbase + parity * 6 * 32;
    for (int j = 0; j < 32; ++j) {
      int   rr = ((const int*)buf)[j];
      int   cc = ((const int*)buf)[32 + j];
      float a0 = ((const float*)buf)[64 + j];
      float a1 = ((const float*)buf)[96 + j];
      float a2 = ((const float*)buf)[128 + j];
      float a3 = ((const float*)buf)[160 + j];
      edge_wu(rr, cc, a0, a1, a2, a3, lane, uk, wu);
    }
    parity ^= 1;
  }
#else
  for (int tile = wave; tile < NTILES; tile += nwav) {
    const int e = tile * 32 + lane;
    int   r = rows[e], c = cols[e];
    float w0 = wv[e], w1 = wv[GNNZ + e], w2 = wv[2 * GNNZ + e], w3 = wv[3 * GNNZ + e];
    for (int j = 0; j < 32; ++j) {
      int   rr = __shfl(r, j, 32);
      int   cc = __shfl(c, j, 32);
      float a0 = __shfl(w0, j, 32);
      float a1 = __shfl(w1, j, 32);
      float a2 = __shfl(w2, j, 32);
      float a3 = __shfl(w3, j, 32);
      edge_wu(rr, cc, a0, a1, a2, a3, lane, uk, wu);
    }
  }
#endif
}

// ---------------- SpMM pass 1: agg = sum_l W_l @ tmp_l ----------------
__global__ __launch_bounds__(256) void scatter_agg(
    const float* __restrict__ wv, const int* __restrict__ rows,
    const int* __restrict__ cols, const float4* __restrict__ tmp4,
    float* __restrict__ agg) {
  const int lane = threadIdx.x & 31;
  const int wib  = threadIdx.x >> 5;
  const int wave = blockIdx.x * (blockDim.x >> 5) + wib;
  const int nwav = gridDim.x * (blockDim.x >> 5);

#if USE_ASYNC_LDS
  __shared__ unsigned smem[8 * 2 * 6 * 32];   // 8 waves x 2 buffers x 768B
  unsigned* wbase = smem + wib * (2 * 6 * 32);
  int tile = wave;
  int parity = 0;
  if (tile < NTILES) stage_tile(tile, wbase, lane, rows, cols, wv);
  for (; tile < NTILES; tile += nwav) {
    const int nxt = tile + nwav;
    if (nxt < NTILES) {
      stage_tile(nxt, wbase + (parity ^ 1) * 6 * 32, lane, rows, cols, wv);
      wait_asynccnt<6>();
    } else {
      wait_asynccnt<0>();
    }
    asm volatile("" ::: "memory");
    const unsigned* buf = wbase + parity * 6 * 32;
    for (int j = 0; j < 32; ++j) {
      int   rr = ((const int*)buf)[j];
      int   cc = ((const int*)buf)[32 + j];
      float a0 = ((const float*)buf)[64 + j];
      float a1 = ((const float*)buf)[96 + j];
      float a2 = ((const float*)buf)[128 + j];
      float a3 = ((const float*)buf)[160 + j];
      edge_agg(rr, cc, a0, a1, a2, a3, lane, tmp4, agg);
    }
    parity ^= 1;
  }
#else
  for (int tile = wave; tile < NTILES; tile += nwav) {
    const int e = tile * 32 + lane;
    int   r = rows[e], c = cols[e];
    float w0 = wv[e], w1 = wv[GNNZ + e], w2 = wv[2 * GNNZ + e], w3 = wv[3 * GNNZ + e];
    for (int j = 0; j < 32; ++j) {
      int   rr = __shfl(r, j, 32);
      int   cc = __shfl(c, j, 32);
      float a0 = __shfl(w0, j, 32);
      float a1 = __shfl(w1, j, 32);
      float a2 = __shfl(w2, j, 32);
      float a3 = __shfl(w3, j, 32);
      edge_agg(rr, cc, a0, a1, a2, a3, lane, tmp4, agg);
    }
  }
#endif
}

// ---------------- zero fill ----------------
__global__ __launch_bounds__(256) void zero_kernel(float4* __restrict__ p, int n4) {
  int i = blockIdx.x * blockDim.x + threadIdx.x;
  int stride = gridDim.x * blockDim.x;
  for (; i < n4; i += stride) p[i] = make_float4(0.f, 0.f, 0.f, 0.f);
}

// ---------------- Uk = (d + mu2)^-1 * (d*x + agg), float4 over [N,F] -----
__global__ __launch_bounds__(256) void uk_update(
    const float4* __restrict__ x, const float* __restrict__ d,
    const float4* __restrict__ agg, float4* __restrict__ uk, float mu2) {
  int i = blockIdx.x * blockDim.x + threadIdx.x;    // over NF/4
  if (i >= NF4) return;
  const int row = i >> 4;                           // GF/4 = 16 float4 per row
  const float dv  = d[row];
  const float inv = 1.0f / (dv + mu2);
  const float4 xv = x[i];
  const float4 av = agg[i];
  uk[i] = make_float4(inv * (dv * xv.x + av.x), inv * (dv * xv.y + av.y),
                      inv * (dv * xv.z + av.z), inv * (dv * xv.w + av.w));
}

// ---- fused Q/Lambda/tmp update over interleaved [N,F,L]:
// one float4 = the 4 operator values of one (node, feature).
// T holds WU on input, tmp_next on output (in place).
__global__ __launch_bounds__(256) void qlam_update(
    float4* __restrict__ T, float4* __restrict__ Lam,
    const float* __restrict__ d, float mu2, float mu2n) {
  int i = blockIdx.x * blockDim.x + threadIdx.x;    // over LNF/4 == n*GF+f
  if (i >= LNF4) return;
  const int row = i >> 6;                           // GF float4 per node
  const float dv  = d[row];
  const float inv = 1.0f / mu2;
  // eta per operator: nu = {0, 1, 1/4, 1/16}
  const float e0 = 0.0f;
  const float e1 = dv * inv;
  const float e2 = 0.25f * dv * inv;
  const float e3 = 0.0625f * dv * inv;
  const float4 wu  = T[i];
  const float4 lam = Lam[i];

  float xm, q;
  xm = wu.x - lam.x * inv;
  q  = copysignf(fmaxf(fabsf(xm) - e0, 0.0f), xm);
  const float nl0 = lam.x + mu2 * (q - wu.x);
  const float t0  = mu2n * q + nl0;

  xm = wu.y - lam.y * inv;
  q  = copysignf(fmaxf(fabsf(xm) - e1, 0.0f), xm);
  const float nl1 = lam.y + mu2 * (q - wu.y);
  const float t1  = mu2n * q + nl1;

  xm = wu.z - lam.z * inv;
  q  = copysignf(fmaxf(fabsf(xm) - e2, 0.0f), xm);
  const float nl2 = lam.z + mu2 * (q - wu.z);
  const float t2  = mu2n * q + nl2;

  xm = wu.w - lam.w * inv;
  q  = copysignf(fmaxf(fabsf(xm) - e3, 0.0f), xm);
  const float nl3 = lam.w + mu2 * (q - wu.w);
  const float t3  = mu2n * q + nl3;

  Lam[i] = make_float4(nl0, nl1, nl2, nl3);
  T[i]   = make_float4(t0, t1, t2, t3);
}

// ---------------- host orchestration ----------------
extern "C" void kernel_launch(void* const* d_in, const int* in_sizes, int n_in,
                              void* d_out, int out_size, void* d_ws, size_t ws_size,
                              hipStream_t stream) {
  (void)in_sizes; (void)n_in; (void)out_size; (void)ws_size;
  const float* x    = (const float*)d_in[0];   // [N, F]
  const float* wv   = (const float*)d_in[1];   // [L, NNZ]
  const float* d    = (const float*)d_in[2];   // [N]
  const int*   rows = (const int*)d_in[3];     // [NNZ]
  const int*   cols = (const int*)d_in[4];     // [NNZ]
  float* uk = (float*)d_out;                   // [N, F] output, doubles as Uk

  float* T   = (float*)d_ws;                   // [N,F,L]: WU in / tmp_next out
  float* Lam = T + LNF;                        // [N,F,L]
  float* agg = Lam + LNF;                      // [N,F]

  const int SC_BLOCKS = 782;                   // 6256 waves over 25000 tiles
  const int UK_BLOCKS = (NF4 + 255) / 256;     // 3125
  const int QL_BLOCKS = (LNF4 + 255) / 256;    // 12500

  zero_kernel<<<2048, 256, 0, stream>>>((float4*)Lam, LNF4);

  float mu2 = 1.0f;
  for (int it = 0; it < GITERS; ++it) {
    zero_kernel<<<1024, 256, 0, stream>>>((float4*)agg, NF4);
    if (it > 0)
      scatter_agg<<<SC_BLOCKS, 256, 0, stream>>>(wv, rows, cols,
                                                 (const float4*)T, agg);
    uk_update<<<UK_BLOCKS, 256, 0, stream>>>(
        (const float4*)x, d, (const float4*)agg, (float4*)uk, mu2);
    if (it + 1 < GITERS) {
      zero_kernel<<<2048, 256, 0, stream>>>((float4*)T, LNF4);
      scatter_wu<<<SC_BLOCKS, 256, 0, stream>>>(wv, rows, cols, uk, T);
      const float mu2n = fminf(1.1f * mu2, 1.0e6f);
      qlam_update<<<QL_BLOCKS, 256, 0, stream>>>(
          (float4*)T, (float4*)Lam, d, mu2, mu2n);
      mu2 = mu2n;
    }
  }
}